// ALIF_Spike_67422396612612
// MI455X (gfx1250) — compile-verified
//
#include <hip/hip_runtime.h>
#include <cstdint>
#include <cstddef>

// LIF/ALIF spiking scan over t:  mem = mem*DECAY*(1-spk) + x_t ; spk = mem > THRESH
// x: [100, 128, 4096] f32  ->  spikes: [100, 128, 4096] f32
//
// Memory-bound streaming kernel (~420 MB @ 23.3 TB/s -> ~18 us floor).
// CDNA5 async global->LDS data mover (GLOBAL_LOAD_ASYNC_TO_LDS_B128 / ASYNCcnt)
// used as a 12-deep per-wave prefetch pipeline over a 16-stage LDS ring.
// Ring > in-flight depth means a stage is rewritten only 4 iterations after its
// ds_load was consumed, so no explicit DS-counter wait is needed in the loop.

typedef float v4f __attribute__((ext_vector_type(4)));

#define THRESH 0.5f
#define DECAY  0.2f

constexpr int T        = 100;
constexpr int BN       = 128 * 4096;   // floats per timestep plane
constexpr int PV4      = BN / 4;       // 131072 float4 per plane
constexpr int BLOCK    = 256;          // 8 wave32 per block
constexpr int RING     = 16;           // LDS ring stages -> 64 KB per block
constexpr int INFLIGHT = 12;           // outstanding async loads per thread

__device__ __forceinline__ void lif_step(v4f& mem, v4f& spk, v4f xv) {
    mem = mem * DECAY * (1.0f - spk) + xv;   // hard reset (to zero) on spike
#pragma unroll
    for (int i = 0; i < 4; ++i)
        spk[i] = (mem[i] > THRESH) ? 1.0f : 0.0f;
}

__global__ __launch_bounds__(BLOCK)
void alif_spike_kernel(const v4f* __restrict__ x, v4f* __restrict__ out) {
    __shared__ v4f buf[RING * BLOCK];

    const int tid = threadIdx.x;
    const long long g = (long long)blockIdx.x * BLOCK + tid;   // my float4 column
    const v4f* xp = x + g;
    v4f*       op = out + g;

    // LDS byte address of my stage-0 slot (generic LDS pointer: low 32 bits are
    // the wave-relative LDS byte offset; hardware adds LDS_BASE).
    const unsigned lds0 = (unsigned)(uintptr_t)(&buf[tid]);

    // ---- prologue: fill the pipeline (ASYNCcnt = INFLIGHT) ----
#pragma unroll
    for (int s = 0; s < INFLIGHT; ++s) {
        const unsigned la = lds0 + (unsigned)(s * BLOCK * 16);
        const unsigned long long ga =
            (unsigned long long)(uintptr_t)(xp + (long long)s * PV4);
        asm volatile("global_load_async_to_lds_b128 %0, %1, off th:TH_LOAD_NT"
                     :: "v"(la), "v"(ga) : "memory");
    }

    v4f mem = {0.0f, 0.0f, 0.0f, 0.0f};
    v4f spk = {0.0f, 0.0f, 0.0f, 0.0f};

    // ---- steady state: consume stage t%RING, refill stage (t+INFLIGHT)%RING ----
#pragma unroll 4
    for (int t = 0; t < T - INFLIGHT; ++t) {
        // oldest outstanding async load (time t) has landed in LDS
        asm volatile("s_wait_asynccnt %0" :: "n"(INFLIGHT - 1) : "memory");

        v4f xv = buf[(t & (RING - 1)) * BLOCK + tid];   // ds_load_b128

        {
            const int st = (t + INFLIGHT) & (RING - 1);  // last read 4 iters ago
            const unsigned la = lds0 + (unsigned)(st * BLOCK * 16);
            const unsigned long long ga =
                (unsigned long long)(uintptr_t)(xp + (long long)(t + INFLIGHT) * PV4);
            asm volatile("global_load_async_to_lds_b128 %0, %1, off th:TH_LOAD_NT"
                         :: "v"(la), "v"(ga) : "memory");
        }

        lif_step(mem, spk, xv);
        __builtin_nontemporal_store(spk, op + (long long)t * PV4);  // pure stream
    }

    // ---- drain: last INFLIGHT timesteps already queued/landed ----
    asm volatile("s_wait_asynccnt 0" ::: "memory");
#pragma unroll
    for (int t = T - INFLIGHT; t < T; ++t) {
        v4f xv = buf[(t & (RING - 1)) * BLOCK + tid];
        lif_step(mem, spk, xv);
        __builtin_nontemporal_store(spk, op + (long long)t * PV4);
    }
}

extern "C" void kernel_launch(void* const* d_in, const int* in_sizes, int n_in,
                              void* d_out, int out_size, void* d_ws, size_t ws_size,
                              hipStream_t stream) {
    (void)in_sizes; (void)n_in; (void)out_size; (void)d_ws; (void)ws_size;
    const v4f* x  = (const v4f*)d_in[0];
    v4f*      out = (v4f*)d_out;

    dim3 grid(PV4 / BLOCK);   // 512 blocks
    dim3 block(BLOCK);        // 256 threads = 8 wave32
    hipLaunchKernelGGL(alif_spike_kernel, grid, block, 0, stream, x, out);
}